// MultiCamNoiseCropV2Prompter_39642548142243
// MI455X (gfx1250) — compile-verified
//
#include <hip/hip_runtime.h>
#include <cstdint>

#define IMAGE 224
#define PADSZ 30
#define CROPSZ 164      // IMAGE - 2*PAD
#define NFRAME 16
#define NBATCH 16

// Block = one (b, c, h) row. 224 threads = 7 wave32s.
// Phase 1: build prompt[w] for this row into LDS (shared across 16 frames).
// Phase 2: stream 16 frames x 56 float4 through async global->LDS double buffers.
__global__ __launch_bounds__(224)
void MultiCamNoiseCropV2Prompter_kernel(
    const float* __restrict__ x,
    const float* __restrict__ pu1,  const float* __restrict__ pu10,
    const float* __restrict__ pd1,  const float* __restrict__ pd10,
    const float* __restrict__ pl1,  const float* __restrict__ pl10,
    const float* __restrict__ pr1,  const float* __restrict__ pr10,
    const int*   __restrict__ cam_idx,
    const int*   __restrict__ off_right,
    const int*   __restrict__ off_down,
    float* __restrict__ out)
{
    __shared__ __align__(16) float sprompt[IMAGE];
    __shared__ float4 xbuf[7][2][32];   // per-wave double buffers, 512 B each

    const int tid = threadIdx.x;
    const int h   = blockIdx.x % IMAGE;
    const int c   = (blockIdx.x / IMAGE) % 3;
    const int b   = blockIdx.x / (IMAGE * 3);

    const int ci  = cam_idx[b];
    const int orr = off_right[b];   // in [1, 30]
    const int od  = off_down[b];    // in [1, 30]

    const int off_l = 2 * PADSZ - orr;   // [30, 59]
    const int off_u = 2 * PADSZ - od;    // [30, 59]

    // ---------------- Phase 1: prompt row ----------------
    {
        const int w  = tid;            // 0..223
        const int cc = ci * 3 + c;
        float p;
        if (h < off_u) {                         // top band
            const int n10_u = off_u / 10;
            p = (h < n10_u * 10)
                    ? pu10[(cc * 10 + (h % 10)) * IMAGE + w]
                    : pu1[cc * IMAGE + w];
        } else if (h >= IMAGE - od) {            // bottom band
            const int hd   = h - (IMAGE - od);   // [0, od)
            const int n1_d = od % 10;
            p = (hd < n1_d)
                    ? pd1[cc * IMAGE + w]
                    : pd10[(cc * 10 + ((hd - n1_d) % 10)) * IMAGE + w];
        } else {                                 // middle band
            const int r = h - off_u;             // [0, 163], no clip needed
            if (w < off_l) {                     // left band
                const int n10_l = off_l / 10;
                p = (w < n10_l * 10)
                        ? pl10[(cc * CROPSZ + r) * 10 + (w % 10)]
                        : pl1[cc * CROPSZ + r];
            } else if (w >= IMAGE - orr) {       // right band
                const int wd   = w - (IMAGE - orr);   // [0, orr)
                const int n1_r = orr % 10;
                p = (wd < n1_r)
                        ? pr1[cc * CROPSZ + r]
                        : pr10[(cc * CROPSZ + r) * 10 + ((wd - n1_r) % 10)];
            } else {
                p = 0.0f;                        // interior
            }
        }
        sprompt[w] = p;
    }
    __syncthreads();

    // ---------------- Phase 2: stream 16 frames ----------------
    const int wave = tid >> 5;     // 0..6
    const int lane = tid & 31;

    const size_t base = (size_t)(b * 3 + c) * (NFRAME * IMAGE * IMAGE)
                        + (size_t)h * IMAGE;
    const float* xb = x   + base;
    float*       ob = out + base;

    // Item g = wave*128 + it*32 + lane, g in [0, 896): t = g/56, w4 = g%56.
    int t_[4], w4_[4];
#pragma unroll
    for (int it = 0; it < 4; ++it) {
        const int g = wave * 128 + it * 32 + lane;
        t_[it]  = g / 56;
        w4_[it] = g - t_[it] * 56;
    }

    const uint32_t lds0 = (uint32_t)(uintptr_t)&xbuf[wave][0][lane];
    const uint32_t lds1 = (uint32_t)(uintptr_t)&xbuf[wave][1][lane];

    auto issue = [&](int it, uint32_t la) {
        const uint64_t ga = (uint64_t)(uintptr_t)(xb + (size_t)t_[it] * (IMAGE * IMAGE)
                                                     + w4_[it] * 4);
        // HBM -> LDS, 16 B per lane, tracked by ASYNCcnt (CDNA5 async path).
        asm volatile("global_load_async_to_lds_b128 %0, %1, off"
                     :: "v"(la), "v"(ga) : "memory");
    };

    auto process = [&](int it, int sel) {
        float4 v = xbuf[wave][sel][lane];
        const float4 p = reinterpret_cast<const float4*>(sprompt)[w4_[it]];
        v.x += p.x; v.y += p.y; v.z += p.z; v.w += p.w;
        *reinterpret_cast<float4*>(ob + (size_t)t_[it] * (IMAGE * IMAGE)
                                      + w4_[it] * 4) = v;
    };

    // Double-buffered pipeline: async loads in flight while adding + storing.
    issue(0, lds0);
    issue(1, lds1);
    asm volatile("s_wait_asynccnt 0x1" ::: "memory");  // buffer 0 ready
    process(0, 0);
    issue(2, lds0);                                    // ds_load(0) already drained by store's dscnt wait
    asm volatile("s_wait_asynccnt 0x1" ::: "memory");  // buffer 1 ready
    process(1, 1);
    issue(3, lds1);
    asm volatile("s_wait_asynccnt 0x1" ::: "memory");  // buffer 0 (iter 2) ready
    process(2, 0);
    asm volatile("s_wait_asynccnt 0x0" ::: "memory");  // buffer 1 (iter 3) ready
    process(3, 1);
}

extern "C" void kernel_launch(void* const* d_in, const int* in_sizes, int n_in,
                              void* d_out, int out_size, void* d_ws, size_t ws_size,
                              hipStream_t stream) {
    const float* x    = (const float*)d_in[0];
    const float* pu1  = (const float*)d_in[1];
    const float* pu10 = (const float*)d_in[2];
    const float* pd1  = (const float*)d_in[3];
    const float* pd10 = (const float*)d_in[4];
    const float* pl1  = (const float*)d_in[5];
    const float* pl10 = (const float*)d_in[6];
    const float* pr1  = (const float*)d_in[7];
    const float* pr10 = (const float*)d_in[8];
    const int*   ci   = (const int*)d_in[9];
    const int*   orr  = (const int*)d_in[10];
    const int*   od   = (const int*)d_in[11];
    float* out = (float*)d_out;

    dim3 grid(NBATCH * 3 * IMAGE);   // 10752 blocks: one per (b, c, h) row
    dim3 block(IMAGE);               // 224 threads = 7 wave32s

    MultiCamNoiseCropV2Prompter_kernel<<<grid, block, 0, stream>>>(
        x, pu1, pu10, pd1, pd10, pl1, pl10, pr1, pr10, ci, orr, od, out);
}